// NonlinearIB_58531814310095
// MI455X (gfx1250) — compile-verified
//
#include <hip/hip_runtime.h>
#include <hip/hip_bf16.h>
#include <math.h>

typedef __attribute__((ext_vector_type(2))) float v2f;
typedef __attribute__((ext_vector_type(8))) float v8f;

#define NROWS 8192
#define NDIM  64
#define COLS_PER_BLOCK 128
#define LDS_STRIDE 66          // 64 + 2 pad: breaks 256B bank aliasing for ds_load_b64
#define BETA 0.01f
#define TWO_PI 6.28318530717958647692f

// ---------------- Kernel 1: y = x + exp(0.5*nv) * noise ----------------
__global__ __launch_bounds__(256) void k_y(const float* __restrict__ x,
                                           const float* __restrict__ noise,
                                           const float* __restrict__ nvp,
                                           float* __restrict__ y, int n) {
    int i = blockIdx.x * 256 + threadIdx.x;
    float s = __expf(0.5f * nvp[0]);
    if (i < n) y[i] = fmaf(s, noise[i], x[i]);
}

// ---------------- Kernel 2: sq[i] = sum_k x[i,k]^2 ----------------
__global__ __launch_bounds__(256) void k_sq(const float* __restrict__ x,
                                            float* __restrict__ sq) {
    int row = blockIdx.x * 256 + threadIdx.x;
    const float4* r = (const float4*)(x + (size_t)row * NDIM);
    float s = 0.0f;
#pragma unroll
    for (int j = 0; j < NDIM / 4; ++j) {
        float4 v = r[j];
        s = fmaf(v.x, v.x, s); s = fmaf(v.y, v.y, s);
        s = fmaf(v.z, v.z, s); s = fmaf(v.w, v.w, s);
    }
    sq[row] = s;
}

// ---------------- Kernel 3: fused Gram-tile + dual online logsumexp ----------------
// grid = NROWS/128 workgroups of 256 threads (8 waves). Wave w handles rows
// [blk*128 + w*16, +16). Columns streamed in 128-wide LDS-staged blocks
// (register double-buffered); each 16x16 Gram tile built from 16 chained
// V_WMMA_F32_16X16X4_F32 with all B fragments preloaded from LDS.
__global__ __launch_bounds__(256) void k_pairwise(const float* __restrict__ x,
                                                  const float* __restrict__ sq,
                                                  const float* __restrict__ kvp,
                                                  const float* __restrict__ nvp,
                                                  float* __restrict__ lse1_out,
                                                  float* __restrict__ lse2_out) {
    __shared__ float ldsX[COLS_PER_BLOCK * LDS_STRIDE];
    __shared__ float ldsSq[COLS_PER_BLOCK];

    const int tid      = threadIdx.x;
    const int lane     = tid & 31;
    const int wave     = tid >> 5;
    const int half     = lane >> 4;      // 0: K pair {0,1} mod 4, 1: pair {2,3}
    const int l15      = lane & 15;
    const int rowBase  = blockIdx.x * (8 * 16) + wave * 16;

    const float kv = kvp[0], nv = nvp[0];
    const float total_var = __expf(nv) + __expf(kv);
    const float a1 = 1.0f / (2.0f * total_var);        // MI-bound temperature
    const float a2 = 1.0f / (2.0f * __expf(kv));       // KDE temperature

    // A fragments: 16 chunks of 16x4 f32.
    // a[ch] = { A[M=l15, K=4ch+2*half], A[M=l15, K=4ch+2*half+1] }
    v2f a[16];
    {
        const float* ar = x + (size_t)(rowBase + l15) * NDIM + 2 * half;
#pragma unroll
        for (int ch = 0; ch < 16; ++ch)
            a[ch] = *(const v2f*)(ar + 4 * ch);
    }

    float sqA[8];
#pragma unroll
    for (int k = 0; k < 8; ++k)
        sqA[k] = sq[rowBase + k + 8 * half];

    float l1[8], l2[8], m2[8];
#pragma unroll
    for (int k = 0; k < 8; ++k) { l1[k] = 0.0f; l2[k] = 0.0f; m2[k] = -1e30f; }

    // ---- register double buffer for the 128-col x-block (32KB, contiguous) ----
    float4 stage[8];
    float  stageSq;
#pragma unroll
    for (int i = 0; i < 8; ++i)
        stage[i] = ((const float4*)x)[tid + i * 256];          // block cb=0
    stageSq = (tid < COLS_PER_BLOCK) ? sq[tid] : 0.0f;

    for (int cb = 0; cb < NROWS; cb += COLS_PER_BLOCK) {
        __syncthreads();   // previous compute done reading LDS
#pragma unroll
        for (int i = 0; i < 8; ++i) {
            int idx = tid + i * 256;            // float4 index 0..2047
            int c   = idx >> 4;                 // column in block
            int f4  = idx & 15;
            float* dst = &ldsX[c * LDS_STRIDE + f4 * 4];
            dst[0] = stage[i].x; dst[1] = stage[i].y;
            dst[2] = stage[i].z; dst[3] = stage[i].w;
        }
        if (tid < COLS_PER_BLOCK) ldsSq[tid] = stageSq;
        __syncthreads();

        // prefetch next block while this one is computed
        const int nb = cb + COLS_PER_BLOCK;
        if (nb < NROWS) {
#pragma unroll
            for (int i = 0; i < 8; ++i)
                stage[i] = ((const float4*)(x + (size_t)nb * NDIM))[tid + i * 256];
            if (tid < COLS_PER_BLOCK) stageSq = sq[nb + tid];
        }

        for (int ct = 0; ct < COLS_PER_BLOCK / 16; ++ct) {
            // preload ALL B fragments so the WMMA chain runs without dscnt-0 stalls
            v2f b[16];
            const float* bp = &ldsX[(ct * 16 + l15) * LDS_STRIDE + 2 * half];
#pragma unroll
            for (int ch = 0; ch < 16; ++ch)
                b[ch] = *(const v2f*)(bp + 4 * ch);

            v8f c = {};
#pragma unroll
            for (int ch = 0; ch < 16; ++ch)
                c = __builtin_amdgcn_wmma_f32_16x16x4_f32(
                        /*neg_a=*/false, a[ch], /*neg_b=*/false, b[ch],
                        /*c_mod=*/(short)0, c, /*reuse_a=*/false, /*reuse_b=*/false);

            const float sqB = ldsSq[ct * 16 + l15];
            const int   col = cb + ct * 16 + l15;
#pragma unroll
            for (int k = 0; k < 8; ++k) {
                const int rowG = rowBase + k + 8 * half;
                float d = fmaxf(sqA[k] + sqB - 2.0f * c[k], 0.0f);
                // loss1: diagonal included (exp(0)=1 is the max term) -> plain sum
                l1[k] += __expf(-d * a1);
                // loss2: diagonal masked with finite sentinel; online max-logsumexp
                float s2 = (col == rowG) ? -1e38f : (-d * a2);
                float nm = fmaxf(m2[k], s2);
                l2[k] = l2[k] * __expf(m2[k] - nm) + __expf(s2 - nm);
                m2[k] = nm;
            }
        }
    }

    // Butterfly merge across the 16 column-lanes of each half-wave
#pragma unroll
    for (int k = 0; k < 8; ++k) {
#pragma unroll
        for (int m = 1; m <= 8; m <<= 1) {
            l1[k] += __shfl_xor(l1[k], m, 32);
            float om = __shfl_xor(m2[k], m, 32);
            float ol = __shfl_xor(l2[k], m, 32);
            float nm = fmaxf(m2[k], om);
            l2[k] = l2[k] * __expf(m2[k] - nm) + ol * __expf(om - nm);
            m2[k] = nm;
        }
    }
    if (l15 == 0) {
#pragma unroll
        for (int k = 0; k < 8; ++k) {
            int rowG = rowBase + k + 8 * half;
            lse1_out[rowG] = __logf(l1[k]);
            lse2_out[rowG] = m2[k] + __logf(l2[k]);
        }
    }
}

// ---------------- Kernel 4: deterministic scalar reduction ----------------
__global__ __launch_bounds__(256) void k_final(const float* __restrict__ lse1,
                                               const float* __restrict__ lse2,
                                               const float* __restrict__ kvp,
                                               const float* __restrict__ nvp,
                                               float* __restrict__ out2) {
    __shared__ float s1[256], s2[256];
    float p1 = 0.0f, p2 = 0.0f;
    for (int i = threadIdx.x; i < NROWS; i += 256) { p1 += lse1[i]; p2 += lse2[i]; }
    s1[threadIdx.x] = p1; s2[threadIdx.x] = p2;
    __syncthreads();
    for (int st = 128; st > 0; st >>= 1) {
        if (threadIdx.x < st) {
            s1[threadIdx.x] += s1[threadIdx.x + st];
            s2[threadIdx.x] += s2[threadIdx.x + st];
        }
        __syncthreads();
    }
    if (threadIdx.x == 0) {
        const float kv = kvp[0], nv = nvp[0];
        const float Nf = (float)NROWS, dims = (float)NDIM;
        // MI upper bound
        float total_var = expf(nv) + expf(kv);
        float nc1 = 0.5f * dims * logf(TWO_PI * total_var);
        float mean_lp1 = s1[0] / Nf - logf(Nf) - nc1;
        float h  = -mean_lp1;
        out2[0] = BETA * (h - nc1);
        // KDE leave-one-out
        float kdevar = expf(kv);
        float nc2 = 0.5f * dims * logf(TWO_PI * kdevar);
        float mean_lp2 = s2[0] / Nf - logf(Nf - 1.0f) - nc2;
        out2[1] = -mean_lp2;
    }
}

extern "C" void kernel_launch(void* const* d_in, const int* in_sizes, int n_in,
                              void* d_out, int out_size, void* d_ws, size_t ws_size,
                              hipStream_t stream) {
    const float* x     = (const float*)d_in[0];
    const float* noise = (const float*)d_in[1];
    const float* kv    = (const float*)d_in[2];
    const float* nv    = (const float*)d_in[3];
    float* out = (float*)d_out;
    float* ws  = (float*)d_ws;

    float* sq   = ws;                 // [0, 8192)
    float* lse1 = ws + NROWS;         // [8192, 16384)
    float* lse2 = ws + 2 * NROWS;     // [16384, 24576)

    const int nd = NROWS * NDIM;
    k_y<<<(nd + 255) / 256, 256, 0, stream>>>(x, noise, nv, out, nd);
    k_sq<<<NROWS / 256, 256, 0, stream>>>(x, sq);
    k_pairwise<<<NROWS / 128, 256, 0, stream>>>(x, sq, kv, nv, lse1, lse2);
    k_final<<<1, 256, 0, stream>>>(lse1, lse2, kv, nv, out + nd);
}